// MLPContrastive_41807211659400
// MI455X (gfx1250) — compile-verified
//
#include <hip/hip_runtime.h>

// Problem constants (from reference): N=2048, K=128, D=768, H=512
#define NN 2048
#define KK 128
#define DD 768
#define HH 512

typedef float v2f __attribute__((ext_vector_type(2)));
typedef float v8f __attribute__((ext_vector_type(8)));

// ---------------------------------------------------------------------------
// Async memory->LDS copy of 16 bytes per lane (CDNA5, ASYNCcnt-tracked).
// LDS operand is the raw LDS byte offset (low 32 bits of the flat pointer:
// see ISA 10.2).
// ---------------------------------------------------------------------------
__device__ __forceinline__ void async_copy_b128(const void* gptr, void* lptr) {
    unsigned int lds_off = (unsigned int)(unsigned long long)(uintptr_t)lptr;
    unsigned long long gaddr = (unsigned long long)(uintptr_t)gptr;
    asm volatile("global_load_async_to_lds_b128 %0, %1, off"
                 :: "v"(lds_off), "v"(gaddr)
                 : "memory");
}
__device__ __forceinline__ void wait_async0() {
    asm volatile("s_wait_asynccnt 0x0" ::: "memory");
}

// ---------------------------------------------------------------------------
// C[M,N] = A[M,Kd] @ B[Kd,N] (+ bias[N] if bias != nullptr), all fp32.
// V_WMMA_F32_16X16X4_F32 (wave32). Block = 256 threads = 8 waves (4M x 2N);
// block tile 64(M) x 128(N); wave tile 16 x 64 (4 x v8f accumulators).
//
// B is staged through LDS with double-buffered GLOBAL_LOAD_ASYNC_TO_LDS_B128
// in K-chunks of 8 rows x 128 cols (4 KB/chunk, one b128 per thread per
// stage). A is read directly (each lane owns one M-row, b64 per k-step).
//
// Fragment layouts per CDNA5 ISA 7.12.2:
//   A (16x4): lanes 0-15 -> M=lane, v0=K0,v1=K1; lanes 16-31 -> K2,K3
//   B (4x16): lanes 0-15 -> N=lane, v0=K0,v1=K1; lanes 16-31 -> K2,K3
//   C/D (16x16): VGPR r: lanes 0-15 -> M=r, lanes 16-31 -> M=r+8; N=lane&15
// ---------------------------------------------------------------------------
__global__ __launch_bounds__(256)
void gemm_wmma_f32(const float* __restrict__ A, const float* __restrict__ B,
                   const float* __restrict__ bias, float* __restrict__ C,
                   int M, int Kd, int N)
{
    __shared__ __align__(16) float Bs[2][8][128];   // 8 KB double buffer

    const int tid  = threadIdx.x;
    const int lane = tid & 31;
    const int wave = tid >> 5;
    const int wm   = wave >> 1;          // 0..3
    const int wn   = wave & 1;           // 0..1
    const int kh   = lane >> 4;          // selects K pair {0,1} vs {2,3}
    const int lid  = lane & 15;

    const int m0  = blockIdx.y * 64 + wm * 16;
    const int n0b = blockIdx.x * 128;    // block N origin

    // Per-thread slice of one 8x128 B chunk: row = tid/32, col quad = (tid%32)*4
    const int srow = tid >> 5;
    const int scol = (tid & 31) * 4;

    auto stage = [&](int buf, int k0) {
        const float* g = B + (size_t)(k0 + srow) * N + n0b + scol;
        async_copy_b128(g, &Bs[buf][srow][scol]);
    };

    v8f acc[4] = {};
    const float* ap = A + (size_t)(m0 + lid) * Kd + kh * 2;

    const int steps = Kd >> 3;           // K-chunks of 8
    stage(0, 0);

    for (int i = 0; i < steps; ++i) {
        wait_async0();                   // own async slice complete
        __syncthreads();                 // -> whole chunk visible; also orders
                                         //    next prefetch after last reads
        if (i + 1 < steps) stage((i + 1) & 1, (i + 1) * 8);

        const int k0  = i * 8;
        const int buf = i & 1;
        #pragma unroll
        for (int kk = 0; kk < 8; kk += 4) {
            v2f a;
            a.x = ap[k0 + kk + 0];
            a.y = ap[k0 + kk + 1];

            const int r = kk + kh * 2;
            v2f b[4];
            #pragma unroll
            for (int j = 0; j < 4; ++j) {
                const int col = wn * 64 + j * 16 + lid;
                b[j].x = Bs[buf][r][col];
                b[j].y = Bs[buf][r + 1][col];
            }
            #pragma unroll
            for (int j = 0; j < 4; ++j) {
                acc[j] = __builtin_amdgcn_wmma_f32_16x16x4_f32(
                    false, a, false, b[j], (short)0, acc[j], false, false);
            }
        }
    }

    const int n0 = n0b + wn * 64;
    #pragma unroll
    for (int j = 0; j < 4; ++j) {
        const int col = n0 + j * 16 + lid;
        const float badd = bias ? bias[col] : 0.0f;
        #pragma unroll
        for (int r = 0; r < 8; ++r) {
            const int row = m0 + kh * 8 + r;
            C[(size_t)row * N + col] = acc[j][r] + badd;
        }
    }
}

// ---------------------------------------------------------------------------
// sim[n,k] = sum_h relu(img_proj[n,h] + txt_proj[k,h]) * W2[h] + b2
// (b1 pre-folded into txt_proj by the txt GEMM.)
//
// Block = 256 threads = 8 waves; each wave owns one n-row. txt_proj streams
// through LDS in 64-row chunks via GLOBAL_LOAD_ASYNC_TO_LDS_B128; row stride
// padded to 516 floats (2064 B: 16B-aligned for b128, 516 mod 64 = 4 so the
// 32 lanes' k-rows are at worst 2-way bank-conflicted). Each lane owns one
// (n,k) dot over H=512 using float4 LDS reads; img row and W2 are lane-
// uniform broadcasts. No cross-lane reduction.
// Static LDS: 64*516*4 + 8*512*4 + 512*4 = 150528 B (< 320 KB/WGP).
// ---------------------------------------------------------------------------
__global__ __launch_bounds__(256)
void fused_sim(const float* __restrict__ img_proj,   // (NN, HH)
               const float* __restrict__ txt_proj,   // (KK, HH), includes b1
               const float* __restrict__ W2,         // (HH)
               const float* __restrict__ b2,         // (1)
               float* __restrict__ out)              // (NN, KK)
{
    constexpr int KC = 64;            // txt rows per LDS chunk
    constexpr int TS = HH + 4;        // 516: padded, 16B-aligned row stride

    __shared__ __align__(16) float txt_s[KC * TS];
    __shared__ __align__(16) float img_s[8 * HH];
    __shared__ __align__(16) float w2_s[HH];

    const int tid  = threadIdx.x;
    const int lane = tid & 31;
    const int wave = tid >> 5;
    const int n0   = blockIdx.x * 8;
    const int n    = n0 + wave;

    // Stage this block's 8 img rows (1024 b128) and W2 (128 b128) async.
    for (int q = tid; q < 8 * (HH / 4); q += 256) {    // 4 per thread
        const int r  = q >> 7;            // / (HH/4)
        const int hq = (q & (HH / 4 - 1)) * 4;
        async_copy_b128(img_proj + (size_t)(n0 + r) * HH + hq, &img_s[r * HH + hq]);
    }
    if (tid < HH / 4)
        async_copy_b128(W2 + tid * 4, &w2_s[tid * 4]);

    const float bias2 = b2[0];

    for (int kc = 0; kc < KK; kc += KC) {
        // Stage txt chunk: KC rows x 128 b128 = 8192 units, 32 per thread.
        for (int q = tid; q < KC * (HH / 4); q += 256) {
            const int k  = q >> 7;
            const int hq = (q & (HH / 4 - 1)) * 4;
            async_copy_b128(txt_proj + (size_t)(kc + k) * HH + hq,
                            &txt_s[k * TS + hq]);
        }
        wait_async0();
        __syncthreads();

        #pragma unroll
        for (int kk = 0; kk < KC; kk += 32) {
            const float4* trow = (const float4*)&txt_s[(kk + lane) * TS];
            const float4* irow = (const float4*)&img_s[wave * HH];
            const float4* w24  = (const float4*)w2_s;
            float acc = 0.0f;
            #pragma unroll 8
            for (int hq = 0; hq < HH / 4; ++hq) {
                const float4 iv = irow[hq];
                const float4 tv = trow[hq];
                const float4 wv = w24[hq];
                acc = fmaf(fmaxf(iv.x + tv.x, 0.0f), wv.x, acc);
                acc = fmaf(fmaxf(iv.y + tv.y, 0.0f), wv.y, acc);
                acc = fmaf(fmaxf(iv.z + tv.z, 0.0f), wv.z, acc);
                acc = fmaf(fmaxf(iv.w + tv.w, 0.0f), wv.w, acc);
            }
            out[(size_t)n * KK + (kc + kk + lane)] = acc + bias2;
        }
        __syncthreads();   // txt_s reused next chunk
    }
}

// ---------------------------------------------------------------------------
// Inputs (setup_inputs order): image_feat(N,D), text_feat(K,D),
//   W1(2D,H), b1(H), W2(H,1), b2(1). Output: sim(N,K) fp32.
// Workspace: img_proj (N*H f32 = 4 MB) then txt_proj (K*H f32 = 256 KB).
// ---------------------------------------------------------------------------
extern "C" void kernel_launch(void* const* d_in, const int* in_sizes, int n_in,
                              void* d_out, int out_size, void* d_ws, size_t ws_size,
                              hipStream_t stream) {
    (void)in_sizes; (void)n_in; (void)out_size; (void)ws_size;

    const float* image_feat = (const float*)d_in[0];
    const float* text_feat  = (const float*)d_in[1];
    const float* W1         = (const float*)d_in[2];
    const float* b1         = (const float*)d_in[3];
    const float* W2         = (const float*)d_in[4];
    const float* b2         = (const float*)d_in[5];
    float* out = (float*)d_out;

    float* img_proj = (float*)d_ws;                       // NN*HH floats
    float* txt_proj = img_proj + (size_t)NN * HH;         // KK*HH floats

    dim3 blk(256);

    // img_proj = image_feat @ W1[:D]          (no bias)
    gemm_wmma_f32<<<dim3(HH / 128, NN / 64), blk, 0, stream>>>(
        image_feat, W1, nullptr, img_proj, NN, DD, HH);

    // txt_proj = text_feat @ W1[D:] + b1      (bias folded here)
    gemm_wmma_f32<<<dim3(HH / 128, KK / 64), blk, 0, stream>>>(
        text_feat, W1 + (size_t)DD * HH, b1, txt_proj, KK, DD, HH);

    // sim = relu(img_proj[:,None,:] + txt_proj[None,:,:]) @ W2 + b2
    fused_sim<<<dim3(NN / 8), blk, 0, stream>>>(
        img_proj, txt_proj, W2, b2, out);
}